// LocalSquaredDistanceLayer_14310831030969
// MI455X (gfx1250) — compile-verified
//
#include <hip/hip_runtime.h>
#include <hip/hip_bf16.h>
#include <float.h>
#include <math.h>

typedef __attribute__((ext_vector_type(2))) float v2f;
typedef __attribute__((ext_vector_type(8))) float v8f;

#define B_   16
#define T_   512
#define C_   8
#define S_   64
#define KSZ_ 32
#define TOUT_ 481
#define EPS_ 1e-8f

// Workspace layout (floats):
//  [0   .. 127 ]  xmu   (B*C)
//  [128 .. 255 ]  xinv  (B*C)   = 1 / (std + eps)
//  [256 .. 319 ]  kmu   (S)
//  [320 .. 383 ]  kinv  (S)     = 1 / (std + eps)
//  [384 .. 2431]  kn2   (S*KSZ) = sum_c kern_norm[s,k,c]^2
// Total: 9728 bytes.

__global__ void lsd_x_stats(const float* __restrict__ x, float* __restrict__ ws) {
    __shared__ float ssum[128];
    __shared__ float ssq[128];
    const int bc  = blockIdx.x;       // b*8 + c
    const int b   = bc >> 3;
    const int c   = bc & 7;
    const int tid = threadIdx.x;      // 128 threads
    float s = 0.f, q = 0.f;
    for (int i = 0; i < 4; ++i) {
        const int t = tid + i * 128;
        const float v = x[(b * T_ + t) * C_ + c];
        s += v; q += v * v;
    }
    ssum[tid] = s; ssq[tid] = q;
    __syncthreads();
    for (int off = 64; off > 0; off >>= 1) {
        if (tid < off) { ssum[tid] += ssum[tid + off]; ssq[tid] += ssq[tid + off]; }
        __syncthreads();
    }
    if (tid == 0) {
        const float mean = ssum[0] * (1.f / (float)T_);
        const float var  = fmaxf(ssq[0] * (1.f / (float)T_) - mean * mean, 0.f);
        ws[bc]       = mean;
        ws[128 + bc] = 1.f / (sqrtf(var) + EPS_);   // divide once here, not per wave
    }
}

__global__ void lsd_k_stats(const float* __restrict__ kern, float* __restrict__ ws) {
    __shared__ float ssum[256];
    __shared__ float ssq[256];
    __shared__ float smu, sinv;
    const int sidx = blockIdx.x;      // shape index s
    const int tid  = threadIdx.x;     // 256 threads
    const float v = kern[sidx * 256 + tid];
    ssum[tid] = v; ssq[tid] = v * v;
    __syncthreads();
    for (int off = 128; off > 0; off >>= 1) {
        if (tid < off) { ssum[tid] += ssum[tid + off]; ssq[tid] += ssq[tid + off]; }
        __syncthreads();
    }
    if (tid == 0) {
        const float mean = ssum[0] * (1.f / 256.f);
        const float var  = fmaxf(ssq[0] * (1.f / 256.f) - mean * mean, 0.f);
        const float inv  = 1.f / (sqrtf(var) + EPS_);
        ws[256 + sidx] = mean;
        ws[320 + sidx] = inv;
        smu = mean; sinv = inv;
    }
    __syncthreads();
    if (tid < KSZ_) {
        const float mu  = smu;
        const float inv = sinv;
        float acc = 0.f;
        for (int c = 0; c < C_; ++c) {
            const float q = (kern[sidx * 256 + tid * C_ + c] - mu) * inv;
            acc += q * q;
        }
        ws[384 + sidx * KSZ_ + tid] = acc;
    }
}

// One wave (32 lanes) per (b, k, 16-row t-tile).
// dist[m,n] = ||p_m||^2 - 2 p_m.q_n + ||q_n||^2 ; cross term via 2x WMMA f32 16x16x4.
__global__ __launch_bounds__(32) void lsd_wmma_kernel(const float* __restrict__ x,
                                                      const float* __restrict__ kern,
                                                      const float* __restrict__ ws,
                                                      float* __restrict__ out) {
    const int t0 = blockIdx.x * 16;   // 31 tiles cover 481 rows
    const int k  = blockIdx.y;        // 0..31
    const int b  = blockIdx.z;        // 0..15
    const int l  = threadIdx.x;       // 0..31

    const int ch   = k >> 2;          // channel for this k slot
    const int tau0 = (k & 3) * 8;     // time offset of patch window
    const int base = t0 + tau0;

    __shared__ float zsh[24];         // 23 consecutive normalized samples
    __shared__ float pnsh[16];        // ||p_m||^2 for m = 0..15

    // x stats are wave-uniform (b, ch fixed per wave); inverse precomputed
    const float mux = ws[b * C_ + ch];
    const float inx = ws[128 + b * C_ + ch];

    if (l < 23) {
        int t = base + l;
        if (t > T_ - 1) t = T_ - 1;   // clamp: only padded (invalid) rows hit this
        zsh[l] = (x[(b * T_ + t) * C_ + ch] - mux) * inx;
    }
    __syncthreads();

    const int m  = l & 15;            // A-row / B-col within tile
    const int hi = l >> 4;            // lane-half selects K pair {2hi, 2hi+1}

    if (l < 16) {
        float acc = 0.f;
        for (int c = 0; c < 8; ++c) { const float z = zsh[l + c]; acc += z * z; }
        pnsh[l] = acc;
    }
    __syncthreads();

    // A tile (16x4 f32 per WMMA): V0 = A[m, 2hi], V1 = A[m, 2hi+1]; A[m,c] = zsh[m+c]
    v2f a_lo, a_hi;
    a_lo.x = zsh[m + 2 * hi];
    a_lo.y = zsh[m + 2 * hi + 1];
    a_hi.x = zsh[m + 2 * hi + 4];
    a_hi.y = zsh[m + 2 * hi + 5];

    float pn2[8];
    for (int j = 0; j < 8; ++j) pn2[j] = pnsh[j + 8 * hi];

    float md[8];
    for (int j = 0; j < 8; ++j) md[j] = FLT_MAX;

    const int n = m;                  // B column = shape within s-tile
    for (int st = 0; st < 4; ++st) {
        const int s = st * 16 + n;
        const float muk = ws[256 + s];
        const float ink = ws[320 + s];     // precomputed reciprocal: no v_div chains
        const float* kp = kern + s * 256 + k * 8;
        // B tile (4x16 f32): V0 = B[2hi, n], V1 = B[2hi+1, n]; B[c,n] = q[s0+n, k, c]
        v2f b_lo, b_hi;
        b_lo.x = (kp[2 * hi]     - muk) * ink;
        b_lo.y = (kp[2 * hi + 1] - muk) * ink;
        b_hi.x = (kp[2 * hi + 4] - muk) * ink;
        b_hi.y = (kp[2 * hi + 5] - muk) * ink;
        const float kn2 = ws[384 + s * KSZ_ + k];

        v8f g = {0.f, 0.f, 0.f, 0.f, 0.f, 0.f, 0.f, 0.f};
        // D = A x B + C, chained over K = 8 (two K=4 steps), full fp32
        g = __builtin_amdgcn_wmma_f32_16x16x4_f32(false, a_lo, false, b_lo,
                                                  (short)0, g, false, false);
        g = __builtin_amdgcn_wmma_f32_16x16x4_f32(false, a_hi, false, b_hi,
                                                  (short)0, g, false, false);

        for (int j = 0; j < 8; ++j) {
            const float d = pn2[j] + kn2 - 2.f * g[j];
            md[j] = fminf(md[j], d);
        }
    }

    // min over the 16 N-lanes of each half (xor of bits 0..3 stays within a half)
    for (int mask = 1; mask <= 8; mask <<= 1)
        for (int j = 0; j < 8; ++j)
            md[j] = fminf(md[j], __shfl_xor(md[j], mask, 32));

    if (n == 0) {
        for (int j = 0; j < 8; ++j) {
            const int t = t0 + j + 8 * hi;   // D layout: VGPR j -> M = j + 8*hi
            if (t < TOUT_) out[(b * TOUT_ + t) * KSZ_ + k] = md[j];
        }
    }
}

extern "C" void kernel_launch(void* const* d_in, const int* in_sizes, int n_in,
                              void* d_out, int out_size, void* d_ws, size_t ws_size,
                              hipStream_t stream) {
    (void)in_sizes; (void)n_in; (void)out_size; (void)ws_size;
    const float* x    = (const float*)d_in[0];   // (16, 512, 8)
    const float* kern = (const float*)d_in[1];   // (64, 32, 8)
    float* ws  = (float*)d_ws;                   // ~9.5 KB used
    float* out = (float*)d_out;                  // (16, 481, 32)

    lsd_x_stats<<<dim3(B_ * C_), dim3(128), 0, stream>>>(x, ws);
    lsd_k_stats<<<dim3(S_), dim3(256), 0, stream>>>(kern, ws);
    lsd_wmma_kernel<<<dim3(31, KSZ_, B_), dim3(32), 0, stream>>>(x, kern, ws, out);
}